// KNNAggregationVariants_5540507812260
// MI455X (gfx1250) — compile-verified
//
#include <hip/hip_runtime.h>
#include <math.h>

// Problem constants (from reference): B=2, N=4096, D_IN=D_OUT=256, K=16, H=128
#define BB 2
#define NN 4096
#define DD 256
#define KK 16
#define HH 128
#define HLD (HH + 4)    // padded LDS stride for hidden (132 words, 132%64=4)
#define ALD (DD + 4)    // padded LDS stride for A tile  (260 words, 260%64=4)
                        // matches TDM pad: 4 DWORDs per 256 DWORDs

typedef __attribute__((ext_vector_type(2))) float v2f;
typedef __attribute__((ext_vector_type(8))) float v8f;
typedef __attribute__((ext_vector_type(4))) unsigned int v4u;
typedef __attribute__((ext_vector_type(8))) int v8i;
typedef __attribute__((ext_vector_type(4))) int v4i;

__device__ __forceinline__ v8f wmma_f32_k4(v2f a, v2f b, v8f c) {
  // D(16x16,f32) = A(16x4,f32) * B(4x16,f32) + C
  return __builtin_amdgcn_wmma_f32_16x16x4_f32(
      /*neg_a=*/false, a, /*neg_b=*/false, b,
      /*c_mod=*/(short)0, c, /*reuse_a=*/false, /*reuse_b=*/false);
}

// ---------------------------------------------------------------------------
// TDM: DMA a 16x256 f32 tile (row-major, stride 256) from global into LDS
// offset 0, inserting 4 DWORDs of padding after every 256 DWORDs (one row)
// so the LDS image has row stride 260 words == s_a[16][ALD].
// D# layout per CDNA5 ISA 8.3/8.4 (group0: count/lds_addr/global_addr/type=2;
// group1: data_size=4B, pad_enable, pad_interval=7(256dw), pad_amount=3(4dw),
// tensor_dim=256x16, tile_dim=256x16, dim0_stride=256).
// This toolchain's builtin is the 6-arg form:
//   (uint32x4 g0, int32x8 g1, int32x4 g2, int32x4 g3, int32x8 gx, i32 cpol)
// groups 2/3 (+extra) are unused for a 2-D tensor -> zeros.
// ---------------------------------------------------------------------------
__device__ __forceinline__ void tdm_load_tile_16x256(const float* gsrc) {
  const unsigned long long ga = (unsigned long long)(uintptr_t)gsrc;
  v4u g0;
  g0.x = 1u;                                          // count=1 (valid), user D#
  g0.y = 0u;                                          // lds_addr = 0 (s_a first in LDS)
  g0.z = (unsigned)(ga & 0xffffffffu);                // global_addr[31:0]
  g0.w = (unsigned)((ga >> 32) & 0x01ffffffu)         // global_addr[56:32]
         | (2u << 30);                                // type = 2 ("image")
  v8i g1;
  g1[0] = (int)((2u << 16)        // data_size = 4 bytes
              | (1u << 20)        // pad_enable
              | (7u << 22)        // pad_interval: 256 DWORDs
              | (3u << 25));      // pad_amount:   4 DWORDs
  g1[1] = (int)(256u << 16);      // tensor_dim0[15:0] = 256 (bits 79:48)
  g1[2] = (int)(16u << 16);       // tensor_dim0 hi=0 | tensor_dim1[15:0]=16
  g1[3] = (int)(256u << 16);      // tensor_dim1 hi=0 | tile_dim0 = 256
  g1[4] = 16;                     // tile_dim1 = 16, tile_dim2 = 0
  g1[5] = 256;                    // tensor_dim0_stride[31:0] = 256
  g1[6] = 0;                      // stride hi = 0, tensor_dim1_stride = 0
  g1[7] = 0;
  const v4i z4 = {0, 0, 0, 0};    // groups 2/3 unused (2-D tensor)
  const v8i z8 = {0, 0, 0, 0, 0, 0, 0, 0};
  __builtin_amdgcn_tensor_load_to_lds(g0, g1, z4, z4, z8, /*cpol=*/0);
}

// ---------------------------------------------------------------------------
// Kernel 1: feat = features @ W_feat + b_feat  (8192x256 @ 256x256, f32 WMMA)
// grid 512 blocks x 128 threads (4 waves). One 16-row M-tile per block:
//   TDM loads the 16x256 A tile into padded LDS; each wave computes 4 N-tiles
//   with a single A-frag LDS read (b64) feeding 4 WMMAs per k-step.
// ---------------------------------------------------------------------------
__global__ void __launch_bounds__(128)
feat_gemm_kernel(const float* __restrict__ A,
                 const float* __restrict__ W,
                 const float* __restrict__ bias,
                 float* __restrict__ C) {
  __shared__ float s_a[16][ALD];   // MUST be the only LDS object (TDM -> offset 0)

  const int tid  = threadIdx.x;
  const int lane = tid & 31;
  const int wave = tid >> 5;
  const int m0   = blockIdx.x * 16;

  if (wave == 0) {                                   // one DMA per block
    tdm_load_tile_16x256(A + (size_t)m0 * DD);
    __builtin_amdgcn_s_wait_tensorcnt(0);
  }
  __builtin_prefetch(W + (size_t)wave * 64, 0, 0);   // global_prefetch_b8
  __syncthreads();

  const int row  = lane & 15;                        // A/B/C-frag: M/N = lane%16
  const int koff = (lane < 16) ? 0 : 2;              // K split across lane halves
  const int colb = wave * 64 + (lane & 15);          // wave covers 4 N-tiles

  v8f acc[4] = {};
  for (int kk = 0; kk < DD; kk += 4) {
    const v2f a = *(const v2f*)&s_a[row][kk + koff]; // one b64 LDS read, 4 WMMAs
    const float* w0p = W + (size_t)(kk + koff) * DD;
    const float* w1p = w0p + DD;
    #pragma unroll
    for (int nt = 0; nt < 4; ++nt) {
      v2f b;
      b.x = w0p[colb + nt * 16];
      b.y = w1p[colb + nt * 16];
      acc[nt] = wmma_f32_k4(a, b, acc[nt]);
    }
  }

  const int mtop = m0 + ((lane < 16) ? 0 : 8);       // D: M = v + 8*(lane>=16)
  #pragma unroll
  for (int nt = 0; nt < 4; ++nt) {
    const int col = colb + nt * 16;
    const float bn = bias[col];
    #pragma unroll
    for (int v = 0; v < 8; ++v)
      C[(size_t)(mtop + v) * DD + col] = acc[nt][v] + bn;
  }
}

// ---------------------------------------------------------------------------
// Kernel 2: causal KNN (K=16 smallest dist), one wave32 per query point.
// dist = ||dxy||_2 (safe) + 0.3*|dt|; mask self and future (t_j > t_i).
// Per-lane register top-16 (static-index insertion) + wave argmin merge.
// Invalid entries recorded as idx = -1 (dist = +inf).
// ---------------------------------------------------------------------------
__global__ void knn_kernel(const float* __restrict__ coords,
                           int* __restrict__ knn_idx,
                           float* __restrict__ knn_dist) {
  const float FINF = __builtin_inff();
  const int lane = threadIdx.x & 31;
  const int wave = threadIdx.x >> 5;
  const int q = blockIdx.x * (blockDim.x >> 5) + wave;   // 0 .. B*N-1
  const int b = q >> 12;            // /4096
  const int i = q & (NN - 1);
  const float* cb = coords + (size_t)b * NN * 3;
  const float cxi = cb[i * 3 + 0];
  const float cyi = cb[i * 3 + 1];
  const float czi = cb[i * 3 + 2];

  float bd[KK]; int bi[KK];
  #pragma unroll
  for (int s = 0; s < KK; ++s) { bd[s] = FINF; bi[s] = -1; }

  for (int j = lane; j < NN; j += 32) {
    const float cx = cb[j * 3 + 0];
    const float cy = cb[j * 3 + 1];
    const float cz = cb[j * 3 + 2];
    const float dx = cx - cxi, dy = cy - cyi;
    const float sq = dx * dx + dy * dy;
    const float spatial = (sq > 0.f) ? sqrtf(sq) : 0.f;
    const float d = spatial + 0.3f * fabsf(cz - czi);
    const bool masked = (j == i) || (cz > czi);          // eye | future
    if (!masked &&
        (d < bd[KK - 1] ||
         (d == bd[KK - 1] && (unsigned)j < (unsigned)bi[KK - 1]))) {
      float nd = d; int ni = j;                          // unrolled insertion
      #pragma unroll
      for (int s = 0; s < KK; ++s) {
        const bool less =
            (nd < bd[s]) || (nd == bd[s] && (unsigned)ni < (unsigned)bi[s]);
        const float td = bd[s]; const int ti = bi[s];
        bd[s] = less ? nd : bd[s];
        bi[s] = less ? ni : bi[s];
        nd = less ? td : nd;
        ni = less ? ti : ni;
      }
    }
  }

  // Merge 32 per-lane sorted lists: 16 rounds of wave32 lexicographic argmin.
  const int wq = q * KK;
  for (int r = 0; r < KK; ++r) {
    float v = bd[0]; int ji = bi[0];
    #pragma unroll
    for (int off = 16; off >= 1; off >>= 1) {
      const float ov = __shfl_xor(v, off, 32);
      const int   oj = __shfl_xor(ji, off, 32);
      if (ov < v || (ov == v && (unsigned)oj < (unsigned)ji)) { v = ov; ji = oj; }
    }
    if (lane == 0) {
      knn_idx[wq + r]  = ji;          // -1 when no more valid neighbors
      knn_dist[wq + r] = v;
    }
    if (ji >= 0 && lane == (ji & 31)) {   // winning lane pops its head
      #pragma unroll
      for (int s = 0; s < KK - 1; ++s) { bd[s] = bd[s + 1]; bi[s] = bi[s + 1]; }
      bd[KK - 1] = FINF; bi[KK - 1] = -1;
    }
  }
}

// ---------------------------------------------------------------------------
// Kernel 3: positional-MLP + masked mean aggregation + residual + LayerNorm.
// One block (128 threads = 4 wave32) per query point.
//   hidden(16x128) = relu(rel(16x4) @ W1 + b1)          (VALU, K=4)
//   pe(16x256)     = hidden @ W2 + b2                   (WMMA f32, K=128)
//   out = LN(feat + mean_valid(feat[nbr] + pe)) * gamma + beta
// ---------------------------------------------------------------------------
__global__ void __launch_bounds__(128)
pe_agg_ln_kernel(const float* __restrict__ feat,
                 const float* __restrict__ coords,
                 const int* __restrict__ knn_idx,
                 const float* __restrict__ knn_dist,
                 const float* __restrict__ W1,
                 const float* __restrict__ b1,
                 const float* __restrict__ W2,
                 const float* __restrict__ b2,
                 const float* __restrict__ gamma,
                 const float* __restrict__ beta,
                 float* __restrict__ out) {
  const int q    = blockIdx.x;          // 0 .. B*N-1
  const int b    = q >> 12;
  const int i    = q & (NN - 1);
  const int tid  = threadIdx.x;         // 0..127
  const int lane = tid & 31;
  const int wave = tid >> 5;

  __shared__ float s_rel[KK][4];
  __shared__ int   s_idx[KK];
  __shared__ float s_hidden[KK][HLD];
  __shared__ float s_out[DD];
  __shared__ float s_red[8];

  const float* cb  = coords + (size_t)b * NN * 3;
  const float cxi = cb[i * 3 + 0];
  const float cyi = cb[i * 3 + 1];
  const float czi = cb[i * 3 + 2];

  if (tid < KK) {
    const int   ji = knn_idx[(size_t)q * KK + tid];
    const float d  = knn_dist[(size_t)q * KK + tid];
    s_idx[tid] = ji;
    if (ji >= 0) {
      s_rel[tid][0] = cb[ji * 3 + 0] - cxi;
      s_rel[tid][1] = cb[ji * 3 + 1] - cyi;
      s_rel[tid][2] = cb[ji * 3 + 2] - czi;
      s_rel[tid][3] = d;
    } else {
      s_rel[tid][0] = 0.f; s_rel[tid][1] = 0.f;
      s_rel[tid][2] = 0.f; s_rel[tid][3] = 0.f;
    }
  }
  __syncthreads();

  unsigned vmask = 0; int cnt = 0;
  #pragma unroll
  for (int m = 0; m < KK; ++m)
    if (s_idx[m] >= 0) { vmask |= (1u << m); ++cnt; }

  { // hidden layer: K=4 too small for WMMA -> VALU; h == tid (128 cols)
    const int h = tid;
    const float w0 = W1[0 * HH + h], w1 = W1[1 * HH + h];
    const float w2 = W1[2 * HH + h], w3 = W1[3 * HH + h];
    const float bbv = b1[h];
    #pragma unroll
    for (int m = 0; m < KK; ++m) {
      const float a = bbv + s_rel[m][0] * w0 + s_rel[m][1] * w1 +
                      s_rel[m][2] * w2 + s_rel[m][3] * w3;
      s_hidden[m][h] = fmaxf(a, 0.f);
    }
  }
  __syncthreads();

  const float* feat_b = feat + (size_t)b * NN * DD;
  const float* feat_q = feat_b + (size_t)i * DD;
  const int row   = lane & 15;
  const int koff  = (lane < 16) ? 0 : 2;
  const int mbase = (lane < 16) ? 0 : 8;
  const int colb  = wave * 64 + (lane & 15);       // wave covers 4 N-tiles

  v8f acc[4] = {};
  for (int kk = 0; kk < HH; kk += 4) {
    const v2f a = *(const v2f*)&s_hidden[row][kk + koff];
    const float* w0p = W2 + (size_t)(kk + koff) * DD;
    const float* w1p = w0p + DD;
    #pragma unroll
    for (int nt = 0; nt < 4; ++nt) {
      v2f w;
      w.x = w0p[colb + nt * 16];
      w.y = w1p[colb + nt * 16];
      acc[nt] = wmma_f32_k4(a, w, acc[nt]);
    }
  }

  #pragma unroll
  for (int nt = 0; nt < 4; ++nt) {
    const int col = colb + nt * 16;
    const float bias2 = b2[col];
    float partial = 0.f;                           // masked sum over my 8 rows
    #pragma unroll
    for (int v = 0; v < 8; ++v) {
      const int m = mbase + v;
      if (vmask & (1u << m))
        partial += acc[nt][v] + bias2 + feat_b[(size_t)s_idx[m] * DD + col];
    }
    const float other = __shfl_xor(partial, 16, 32);  // rows 0-7 + rows 8-15
    if (lane < 16) {
      const float total = partial + other;
      const float fq = feat_q[col];
      s_out[col] = (cnt > 0) ? (fq + total / (float)cnt) : fq;
    }
  }
  __syncthreads();

  // LayerNorm over 256 features (each thread owns cols tid, tid+128)
  const float x0 = s_out[tid];
  const float x1 = s_out[tid + HH];
  float s  = x0 + x1;
  float ss = x0 * x0 + x1 * x1;
  #pragma unroll
  for (int off = 16; off >= 1; off >>= 1) {
    s  += __shfl_xor(s, off, 32);
    ss += __shfl_xor(ss, off, 32);
  }
  if (lane == 0) { s_red[wave] = s; s_red[4 + wave] = ss; }
  __syncthreads();
  const float sum   = s_red[0] + s_red[1] + s_red[2] + s_red[3];
  const float sumsq = s_red[4] + s_red[5] + s_red[6] + s_red[7];
  const float mu  = sum * (1.f / DD);
  const float var = sumsq * (1.f / DD) - mu * mu;
  const float inv = rsqrtf(var + 1e-5f);
  float* oq = out + (size_t)q * DD;
  oq[tid]      = (x0 - mu) * inv * gamma[tid] + beta[tid];
  oq[tid + HH] = (x1 - mu) * inv * gamma[tid + HH] + beta[tid + HH];
}

// ---------------------------------------------------------------------------
extern "C" void kernel_launch(void* const* d_in, const int* in_sizes, int n_in,
                              void* d_out, int out_size, void* d_ws, size_t ws_size,
                              hipStream_t stream) {
  (void)in_sizes; (void)n_in; (void)out_size; (void)ws_size;
  const float* features = (const float*)d_in[0];  // (B,N,256)
  const float* coords   = (const float*)d_in[1];  // (B,N,3)
  const float* W_feat   = (const float*)d_in[2];  // (256,256)
  const float* b_feat   = (const float*)d_in[3];  // (256,)
  const float* W_pos1   = (const float*)d_in[4];  // (4,128)
  const float* b_pos1   = (const float*)d_in[5];  // (128,)
  const float* W_pos2   = (const float*)d_in[6];  // (128,256)
  const float* b_pos2   = (const float*)d_in[7];  // (256,)
  const float* gamma    = (const float*)d_in[8];  // (256,)
  const float* beta     = (const float*)d_in[9];  // (256,)
  float* out = (float*)d_out;

  // workspace: feat (8 MB) | knn_idx (512 KB) | knn_dist (512 KB)
  char* ws = (char*)d_ws;
  float* feat     = (float*)ws;
  int*   knn_idx  = (int*)(ws + (size_t)BB * NN * DD * sizeof(float));
  float* knn_dist = (float*)(ws + (size_t)BB * NN * DD * sizeof(float)
                                + (size_t)BB * NN * KK * sizeof(int));

  feat_gemm_kernel<<<(BB * NN) / 16, 128, 0, stream>>>(features, W_feat, b_feat, feat);

  knn_kernel<<<(BB * NN) / 8, 256, 0, stream>>>(coords, knn_idx, knn_dist);

  pe_agg_ln_kernel<<<BB * NN, 128, 0, stream>>>(feat, coords, knn_idx, knn_dist,
                                                W_pos1, b_pos1, W_pos2, b_pos2,
                                                gamma, beta, out);
}